// GridEncoder_59021440581778
// MI455X (gfx1250) — compile-verified
//
#include <hip/hip_runtime.h>
#include <math.h>

// ---------------------------------------------------------------------------
// Instant-NGP multi-resolution hash-grid encoder, forward pass, for MI455X.
// B = 1048576 points, D=3, 16 levels, 2 features/level, out = [B, 32] f32.
//
// Design (memory-bound kernel):
//  * lane -> (point-slot, level): 16 lanes cover the 16 levels of one point,
//    each lane emits one float2 -> wave stores 256 contiguous bytes.
//  * Levels 0-4 dense (no modulo needed: idx < size), levels 5-15 hashed with
//    size == 2^19 exactly -> modulo becomes "& (size-1)". No integer division.
//  * Levels 0+1 (entries [0, 18744), 149,952 B) are staged into LDS once per
//    block via the CDNA5 Tensor Data Mover (tensor_load_to_lds, 6-arg form),
//    waited with s_wait_tensorcnt, then gathered via generic-pointer
//    flat_load_b64 (per-lane LDS/global resolution, no wave divergence).
//  * Per-level constants computed on host in double precision with the exact
//    reference recipe, passed by value.
// ---------------------------------------------------------------------------

#define NLVL          16
#define TPB           1024          // 32 waves / block
#define PTS_PER_BLOCK 1024
#define SLOTS         (TPB / NLVL)  // 64 point-slots per iteration
#define LDS_LEVELS    2
#define LDS_ENTRIES   18744u        // off[2] = 4920 + 13824 (verified on host)

typedef unsigned int uint;
typedef unsigned int u32x4 __attribute__((ext_vector_type(4)));
typedef int          i32x8 __attribute__((ext_vector_type(8)));
typedef int          i32x4 __attribute__((ext_vector_type(4)));

#if defined(__HIP_DEVICE_COMPILE__) && \
    __has_builtin(__builtin_amdgcn_tensor_load_to_lds) && \
    __has_builtin(__builtin_amdgcn_s_wait_tensorcnt)
#define HAVE_TDM 1
#else
#define HAVE_TDM 0
#endif

struct EncParams {
    float    scale[NLVL];
    unsigned side[NLVL];
    unsigned sizem1[NLVL];   // size-1: hash mask (size is 2^19 for hashed lvls)
    unsigned off[NLVL];
    unsigned hashed[NLVL];
    unsigned npts;
};

__device__ __forceinline__ uint umin_u(uint a, uint b) { return a < b ? a : b; }

__launch_bounds__(TPB)
__global__ void grid_encode_kernel(const float* __restrict__ inputs,
                                   const float* __restrict__ emb,
                                   float* __restrict__ out,
                                   EncParams P)
{
    __shared__ float2 lds_emb[LDS_ENTRIES];

#if HAVE_TDM
    if (threadIdx.x < 32u) {                     // wave 0 issues the DMA
        const unsigned long long ga = (unsigned long long)emb;
        // low 32 bits of a generic shared pointer == LDS byte offset
        const uint lds_addr = (uint)(unsigned long long)(&lds_emb[0]);
        const uint NDW = LDS_ENTRIES * 2u;       // 37488 dwords (fits 16 bits)
        u32x4 g0;
        g0[0] = 1u;                              // count=1, user desc, no gather
        g0[1] = lds_addr;                        // lds_addr [63:32]
        g0[2] = (uint)ga;                        // global_addr [95:64]
        g0[3] = ((uint)(ga >> 32) & 0x01FFFFFFu) // global_addr [120:96]
                | 0x80000000u;                   // type=2 ("image") [127:126]
        i32x8 g1;
        g1[0] = (int)(2u << 16);                 // wg_mask=0, data_size=4B
        g1[1] = (int)((NDW & 0xFFFFu) << 16);    // tensor_dim0 lo16 @ [63:48]
        g1[2] = (int)(((NDW >> 16) & 0xFFFFu)    // tensor_dim0 hi16 @ [79:64]
                | (1u << 16));                   // tensor_dim1=1   @ [111:80]
        g1[3] = (int)((NDW & 0xFFFFu) << 16);    // tile_dim0       @ [127:112]
        g1[4] = (int)1;                          // tile_dim1=1, tile_dim2=0
        g1[5] = (int)NDW;                        // tensor_dim0_stride lo32
        g1[6] = 0;
        g1[7] = 0;
        i32x4 gz  = {0, 0, 0, 0};                // groups 2/3 unused (<=2D)
        i32x8 g4z = {0, 0, 0, 0, 0, 0, 0, 0};    // extra group (6-arg form)
        __builtin_amdgcn_tensor_load_to_lds(g0, g1, gz, gz, g4z, 0);
        __builtin_amdgcn_s_wait_tensorcnt(0);
    }
#else
    {   // fallback: cooperative copy
        const float2* ge = (const float2*)emb;
        for (uint i = threadIdx.x; i < LDS_ENTRIES; i += TPB)
            lds_emb[i] = ge[i];
    }
#endif
    __syncthreads();

    const uint tid  = threadIdx.x;
    const uint lvl  = tid & 15u;
    const uint slot = tid >> 4;                  // 0..63

    const float scl    = P.scale[lvl];
    const uint  side   = P.side[lvl];
    const uint  off    = P.off[lvl];
    const uint  mask   = P.sizem1[lvl];
    const bool  hashed = P.hashed[lvl] != 0u;
    const bool  in_lds = lvl < LDS_LEVELS;
    const uint  mulY   = hashed ? 2654435761u : side;
    const uint  mulZ   = hashed ? 805459861u  : side * side;
    const uint  smax   = side - 1u;

    const float2* gptr = (const float2*)emb;
    const float2* lptr = (const float2*)&lds_emb[0];
    float2*       out2 = (float2*)out;

    const uint base = blockIdx.x * PTS_PER_BLOCK;

    #pragma unroll 2
    for (uint it = 0; it < PTS_PER_BLOCK; it += SLOTS) {
        const uint pt = base + it + slot;
        if (pt >= P.npts) break;

        const float* ip = inputs + (size_t)pt * 3u;
        // streaming hint for a future iteration (global_prefetch_b8;
        // speculative prefetch is dropped on translation failure -> OOB safe)
        __builtin_prefetch(ip + 3u * SLOTS, 0, 3);

        const float x = ip[0], y = ip[1], z = ip[2];
        const float px = fmaf(x, scl, 0.5f);
        const float py = fmaf(y, scl, 0.5f);
        const float pz = fmaf(z, scl, 0.5f);
        const float fx = floorf(px), fy = floorf(py), fz = floorf(pz);
        const float tx = px - fx,    ty = py - fy,    tz = pz - fz;
        const uint  ix = (uint)fx,   iy = (uint)fy,   iz = (uint)fz;

        const uint X[2]  = { umin_u(ix, smax),        umin_u(ix + 1u, smax) };
        const uint TY[2] = { umin_u(iy, smax) * mulY, umin_u(iy + 1u, smax) * mulY };
        const uint TZ[2] = { umin_u(iz, smax) * mulZ, umin_u(iz + 1u, smax) * mulZ };
        const float WX[2] = { 1.0f - tx, tx };
        const float WY[2] = { 1.0f - ty, ty };
        const float WZ[2] = { 1.0f - tz, tz };

        float ax = 0.0f, ay = 0.0f;
        #pragma unroll
        for (int cz = 0; cz < 2; ++cz) {
            #pragma unroll
            for (int cy = 0; cy < 2; ++cy) {
                const float wzy = WZ[cz] * WY[cy];
                const uint  hzy = TZ[cz] ^ TY[cy];   // xor-combine (hashed)
                const uint  dzy = TZ[cz] + TY[cy];   // add-combine (dense)
                #pragma unroll
                for (int cx = 0; cx < 2; ++cx) {
                    const uint idx = (hashed ? ((X[cx] ^ hzy) & mask)
                                             : (X[cx] + dzy)) + off;
                    // generic-pointer select -> flat_load_b64 resolves
                    // LDS vs global per lane without wave divergence
                    const float2* sp = in_lds ? (lptr + idx) : (gptr + idx);
                    const float2 f = *sp;
                    const float  w = wzy * WX[cx];
                    ax = fmaf(w, f.x, ax);
                    ay = fmaf(w, f.y, ay);
                }
            }
        }
        out2[(size_t)pt * 16u + lvl] = make_float2(ax, ay);
    }
}

extern "C" void kernel_launch(void* const* d_in, const int* in_sizes, int n_in,
                              void* d_out, int out_size, void* d_ws, size_t ws_size,
                              hipStream_t stream)
{
    (void)n_in; (void)out_size; (void)d_ws; (void)ws_size;
    const float* inputs = (const float*)d_in[0];
    const float* emb    = (const float*)d_in[1];
    float*       out    = (float*)d_out;
    const unsigned npts = (unsigned)(in_sizes[0] / 3);

    // Per-level constants: exact double-precision mirror of the reference.
    EncParams P;
    const double PLS = exp2(log2(2048.0 / 16.0) / 15.0);   // 2^(7/15)
    unsigned long long off = 0;
    for (int l = 0; l < NLVL; ++l) {
        const double s = 16.0 * pow(PLS, (double)l);
        const unsigned side = (unsigned)ceil(s) + 1u;
        const unsigned long long cube = (unsigned long long)side * side * side;
        unsigned long long params = cube < (1ull << 19) ? cube : (1ull << 19);
        params = ((params + 7ull) / 8ull) * 8ull;          // align to 8
        P.scale[l]  = (float)(s - 1.0);
        P.side[l]   = side;
        P.sizem1[l] = (unsigned)params - 1u;
        P.off[l]    = (unsigned)off;
        P.hashed[l] = (cube > params) ? 1u : 0u;
        off += params;
    }
    P.npts = npts;

    const unsigned blocks = (npts + PTS_PER_BLOCK - 1) / PTS_PER_BLOCK;
    grid_encode_kernel<<<blocks, TPB, 0, stream>>>(inputs, emb, out, P);
}